// CatNMFLoss_59691455480500
// MI455X (gfx1250) — compile-verified
//
#include <hip/hip_runtime.h>
#include <hip/hip_bf16.h>
#include <math.h>

// ---------------------------------------------------------------------------
// Fused SSIM + MSE + BCE loss for gfx1250 (MI455X, wave32, WGP, WMMA).
//
//   Kernel 1: one workgroup (128 thr = 4 wave32) per 32x32 tile. Stage 42x42
//             halo of x,y in LDS, horizontal 11-tap separable-Gaussian pass
//             into 5 LDS planes, vertical pass as banded-Toeplitz matmul via
//             V_WMMA_F32_16X16X4_F32 (7 K-chunks of 4, one wave per 16x16
//             quadrant), SSIM formula on position-aligned accumulators,
//             deterministic block reduce -> partials in d_ws. MSE fused into
//             the halo-load loop. All bounds handling via zero-padded LDS
//             (2 pad rows per hconv plane + padded Gaussian band table) so
//             the WMMA inner loop is branch-free straight-line code.
//   Kernel 2: BCE partials. Kernel 3: fixed-order final reduce + weighting.
// ---------------------------------------------------------------------------

typedef __attribute__((ext_vector_type(2))) float v2f;
typedef __attribute__((ext_vector_type(8))) float v8f;

#define IMG_H 512
#define IMG_W 512
#define NIMG  48             // B*C
#define TW    32             // tile width/height (output)
#define HALO  42             // TW + 10 (11-tap halo)
#define HROWS 44             // HALO + 2 zero pad rows (K-chunks reach row 43)
#define XSTR  43             // LDS stride for x/y tiles
#define HSTR  33             // LDS stride for hconv planes
#define PLSZ  (HROWS*HSTR)   // one hconv plane (incl. pad rows)
#define NBLK  (NIMG*16*16)   // 12288 tiles
#define NBCE_BLK 64
#define C1F 0.0001f          // 0.01^2
#define C2F 0.0009f          // 0.03^2

// ---------------------------------------------------------------------------
// Kernel 1: fused SSIM map + MSE per 32x32 tile.
// ---------------------------------------------------------------------------
__global__ __launch_bounds__(128)
void ssim_mse_kernel(const float* __restrict__ x,   // wmed
                     const float* __restrict__ y,   // cover
                     float* __restrict__ ssim_part, // [NBLK]
                     float* __restrict__ mse_part)  // [NBLK]
{
    __shared__ float xs[HALO * XSTR];
    __shared__ float ys[HALO * XSTR];
    __shared__ float hc[5 * PLSZ];   // hpass planes: mx,my,E[xx],E[yy],E[xy]
    __shared__ float gn[12];         // normalized 1D Gaussian (11 taps)
    __shared__ float gpad[48];       // zero-padded band table: gpad[15+t]=gn[t]
    __shared__ float red[128];

    const int tid = threadIdx.x;

    // Zero-padded Gaussian band table (sigma=1.5; gn ⊗ gn == reference win).
    if (tid < 48) gpad[tid] = 0.f;
    if (tid == 0) {
        float g[11]; float s = 0.f;
        for (int i = 0; i < 11; ++i) {
            float d = (float)(i - 5);
            g[i] = expf(-d * d / (2.f * 1.5f * 1.5f));
            s += g[i];
        }
        for (int i = 0; i < 11; ++i) { gn[i] = g[i] / s; gpad[15 + i] = g[i] / s; }
    }
    __syncthreads();

    // ---- Stage halo tile (zero-padded like lax.conv) + fused MSE ----------
    const int bc = blockIdx.z;                 // image plane 0..47
    const int r0 = (int)blockIdx.y * TW - 5;   // halo origin (image coords)
    const int c0 = (int)blockIdx.x * TW - 5;
    const float* __restrict__ xb = x + (size_t)bc * (IMG_H * IMG_W);
    const float* __restrict__ yb = y + (size_t)bc * (IMG_H * IMG_W);

    float msum = 0.f;
    for (int i = tid; i < HALO * HALO; i += 128) {
        int r = i / HALO, c = i - r * HALO;
        int ir = r0 + r, ic = c0 + c;
        bool inb = ((unsigned)ir < (unsigned)IMG_H) & ((unsigned)ic < (unsigned)IMG_W);
        float xv = inb ? xb[(size_t)ir * IMG_W + ic] : 0.f;
        float yv = inb ? yb[(size_t)ir * IMG_W + ic] : 0.f;
        xs[r * XSTR + c] = xv;
        ys[r * XSTR + c] = yv;
        // interior 32x32 pixels partition the image exactly -> fuse MSE here
        if (r >= 5 && r < 5 + TW && c >= 5 && c < 5 + TW) {
            float d = xv - yv;
            msum += d * d;
        }
    }

    // Zero the 2 pad rows of every hconv plane (rows 42,43).
    for (int i = tid; i < 5 * 2 * HSTR; i += 128) {
        int p = i / (2 * HSTR);
        int o = i - p * (2 * HSTR);
        hc[p * PLSZ + HALO * HSTR + o] = 0.f;
    }
    __syncthreads();

    // ---- Horizontal 11-tap pass for the 5 planes --------------------------
    for (int i = tid; i < HALO * TW; i += 128) {
        int r = i >> 5;        // /32
        int c = i & 31;
        float sx = 0.f, sy = 0.f, sxx = 0.f, syy = 0.f, sxy = 0.f;
#pragma unroll
        for (int t = 0; t < 11; ++t) {
            float w  = gn[t];
            float xv = xs[r * XSTR + c + t];
            float yv = ys[r * XSTR + c + t];
            sx  += w * xv;       sy  += w * yv;
            sxx += w * xv * xv;  syy += w * yv * yv;  sxy += w * xv * yv;
        }
        int o = r * HSTR + c;
        hc[0 * PLSZ + o] = sx;
        hc[1 * PLSZ + o] = sy;
        hc[2 * PLSZ + o] = sxx;
        hc[3 * PLSZ + o] = syy;
        hc[4 * PLSZ + o] = sxy;
    }
    __syncthreads();

    // ---- Vertical pass as WMMA: Out(16x16) = W(16x28 banded) x H(28x16) ---
    // A 16x4 f32 layout (ISA 7.12.2): lane l -> M = l&15; VGPR v holds
    // K = v + 2*(l>=16).  B 4x16 row-striped across lanes, same K pattern.
    // C/D: VGPR r -> M = r + 8*(l>=16), N = l&15.  All reads are in-bounds
    // by construction (pad rows + padded band table) -> branch-free.
    const int lane = tid & 31;
    const int wave = tid >> 5;
    const int half = lane >> 4;
    const int m16  = lane & 15;
    const int qr   = (wave >> 1) * 16;   // quadrant row offset in tile
    const int qc   = (wave & 1) * 16;    // quadrant col offset in tile

    v8f a0 = {}, a1 = {}, a2 = {}, a3 = {}, a4 = {};

    // Per-lane invariant pieces of the addressing.
    const int bandBase = 15 + 2 * half - m16;       // gpad index for J=4*kk(+1)
    const int bBase    = (qr + 2 * half) * HSTR + qc + m16;

#pragma unroll
    for (int kk = 0; kk < 7; ++kk) {
        v2f av;
        av.x = gpad[bandBase + 4 * kk];
        av.y = gpad[bandBase + 4 * kk + 1];
        const int o0 = bBase + 4 * kk * HSTR;       // row qr + 4kk + 2*half
        const int o1 = o0 + HSTR;                   // row ... + 1

#if __has_builtin(__builtin_amdgcn_wmma_f32_16x16x4_f32)
        {
            v2f bv; bv.x = hc[0 * PLSZ + o0]; bv.y = hc[0 * PLSZ + o1];
            a0 = __builtin_amdgcn_wmma_f32_16x16x4_f32(false, av, false, bv, (short)0, a0, false, false);
        }
        {
            v2f bv; bv.x = hc[1 * PLSZ + o0]; bv.y = hc[1 * PLSZ + o1];
            a1 = __builtin_amdgcn_wmma_f32_16x16x4_f32(false, av, false, bv, (short)0, a1, false, false);
        }
        {
            v2f bv; bv.x = hc[2 * PLSZ + o0]; bv.y = hc[2 * PLSZ + o1];
            a2 = __builtin_amdgcn_wmma_f32_16x16x4_f32(false, av, false, bv, (short)0, a2, false, false);
        }
        {
            v2f bv; bv.x = hc[3 * PLSZ + o0]; bv.y = hc[3 * PLSZ + o1];
            a3 = __builtin_amdgcn_wmma_f32_16x16x4_f32(false, av, false, bv, (short)0, a3, false, false);
        }
        {
            v2f bv; bv.x = hc[4 * PLSZ + o0]; bv.y = hc[4 * PLSZ + o1];
            a4 = __builtin_amdgcn_wmma_f32_16x16x4_f32(false, av, false, bv, (short)0, a4, false, false);
        }
#else
        // Scalar fallback producing the identical C-layout accumulators.
        (void)av; (void)o0; (void)o1;
#pragma unroll
        for (int r8 = 0; r8 < 8; ++r8) {
            int m = r8 + 8 * half;
#pragma unroll
            for (int jj = 0; jj < 4; ++jj) {
                int J = 4 * kk + jj;
                float a = gpad[15 + J - m];
                int ob = (qr + J) * HSTR + qc + m16;
                a0[r8] += a * hc[0 * PLSZ + ob];
                a1[r8] += a * hc[1 * PLSZ + ob];
                a2[r8] += a * hc[2 * PLSZ + ob];
                a3[r8] += a * hc[3 * PLSZ + ob];
                a4[r8] += a * hc[4 * PLSZ + ob];
            }
        }
#endif
    }

    // ---- SSIM formula: the 5 accumulators are position-aligned ------------
    float ssum = 0.f;
#pragma unroll
    for (int r = 0; r < 8; ++r) {
        float mx  = a0[r], my = a1[r];
        float vx  = a2[r] - mx * mx;
        float vy  = a3[r] - my * my;
        float cv  = a4[r] - mx * my;
        float num = (2.f * mx * my + C1F) * (2.f * cv + C2F);
        float den = (mx * mx + my * my + C1F) * (vx + vy + C2F);
        ssum += num / den;
    }

    // ---- Deterministic block reductions -> per-tile partials --------------
    const int bid = ((int)blockIdx.z * (int)gridDim.y + (int)blockIdx.y) * (int)gridDim.x
                    + (int)blockIdx.x;

    red[tid] = ssum;
    __syncthreads();
#pragma unroll
    for (int s = 64; s > 0; s >>= 1) {
        if (tid < s) red[tid] += red[tid + s];
        __syncthreads();
    }
    if (tid == 0) ssim_part[bid] = red[0];
    __syncthreads();

    red[tid] = msum;
    __syncthreads();
#pragma unroll
    for (int s = 64; s > 0; s >>= 1) {
        if (tid < s) red[tid] += red[tid + s];
        __syncthreads();
    }
    if (tid == 0) mse_part[bid] = red[0];
}

// ---------------------------------------------------------------------------
// Kernel 2: BCE partials. 64 blocks x 256 threads == 16384 elements exactly.
// ---------------------------------------------------------------------------
__global__ __launch_bounds__(256)
void bce_kernel(const float* __restrict__ o, const float* __restrict__ e,
                float* __restrict__ part)
{
    __shared__ float red[256];
    const int tid = threadIdx.x;
    const int idx = (int)blockIdx.x * 256 + tid;
    float ov = o[idx];
    float ev = e[idx];
    float s = -(ov * logf(ev) + (1.f - ov) * logf(1.f - ev));
    red[tid] = s;
    __syncthreads();
#pragma unroll
    for (int st = 128; st > 0; st >>= 1) {
        if (tid < st) red[tid] += red[tid + st];
        __syncthreads();
    }
    if (tid == 0) part[blockIdx.x] = red[0];
}

// ---------------------------------------------------------------------------
// Kernel 3: fixed-order final reduction + curriculum weighting.
// ---------------------------------------------------------------------------
__global__ __launch_bounds__(256)
void finalize_kernel(const float* __restrict__ ws, const int* __restrict__ epoch_p,
                     float* __restrict__ out)
{
    __shared__ float rs[256], rm[256], rb[256];
    const int t = threadIdx.x;
    float ss = 0.f, sm = 0.f, sb = 0.f;
    for (int i = t; i < NBLK; i += 256) {
        ss += ws[i];
        sm += ws[NBLK + i];
    }
    if (t < NBCE_BLK) sb = ws[2 * NBLK + t];
    rs[t] = ss; rm[t] = sm; rb[t] = sb;
    __syncthreads();
#pragma unroll
    for (int s = 128; s > 0; s >>= 1) {
        if (t < s) { rs[t] += rs[t + s]; rm[t] += rm[t + s]; rb[t] += rb[t + s]; }
        __syncthreads();
    }
    if (t == 0) {
        const float inv = 1.f / ((float)NIMG * (float)IMG_H * (float)IMG_W);
        float sv = rs[0] * inv;
        float ml = rm[0] * inv;
        float wl = rb[0] / (16.f * 1024.f);
        int epoch = epoch_p[0];
        float w_img, w_ssim;
        if (epoch <= 12) {
            w_img = 0.05f; w_ssim = 0.05f;
        } else {
            float prog = fminf(1.f, (float)(epoch - 12) / 10.f);
            w_img  = 0.05f + (0.5f - 0.05f) * prog;
            w_ssim = 0.05f + (0.8f - 0.05f) * prog;
        }
        float total = w_img * ml + w_ssim * (1.f - sv) + 3.f * wl;
        out[0] = total;
        out[1] = ml;
        out[2] = sv;
        out[3] = wl;
    }
}

// ---------------------------------------------------------------------------
extern "C" void kernel_launch(void* const* d_in, const int* in_sizes, int n_in,
                              void* d_out, int out_size, void* d_ws, size_t ws_size,
                              hipStream_t stream)
{
    const float* cover   = (const float*)d_in[0];
    const float* wmed    = (const float*)d_in[1];
    const float* wm_orig = (const float*)d_in[2];
    const float* wm_ext  = (const float*)d_in[3];
    const int*   epoch   = (const int*)d_in[4];
    float* out = (float*)d_out;
    float* ws  = (float*)d_ws;   // [0,NBLK): ssim | [NBLK,2N): mse | [2N,2N+64): bce

    dim3 grid(16, 16, NIMG);
    dim3 block(128);
    // diff_ssim(x=wmed, y=cover)
    ssim_mse_kernel<<<grid, block, 0, stream>>>(wmed, cover, ws, ws + NBLK);
    bce_kernel<<<NBCE_BLK, 256, 0, stream>>>(wm_orig, wm_ext, ws + 2 * NBLK);
    finalize_kernel<<<1, 256, 0, stream>>>(ws, epoch, out);
}